// TDTLoss_40638980555327
// MI455X (gfx1250) — compile-verified
//
#include <hip/hip_runtime.h>
#include <hip/hip_bf16.h>
#include <stdint.h>

// ---------------------------------------------------------------------------
// TDT (token-and-duration transducer) loss, MI455X / gfx1250.
//   B=8, T=256, U=64, V=512 (BLANK=512), D=4 durations {1,2,3,4}, SIGMA=0.05
// Phase 1: row softmax stats (HBM-bound, 0.27 GB) with async global->LDS
//          double buffering (CDNA5 ASYNCcnt path).
// Phase 2: per-batch DP over T steps, register-pipelined operands.
// Phase 3: mean of -log-likelihood.
// ---------------------------------------------------------------------------

#define NEGF (-1e30f)
constexpr int   B_    = 8;
constexpr int   T_    = 256;
constexpr int   U_    = 64;
constexpr int   U1_   = 65;
constexpr int   VV    = 513;           // V + 1, BLANK = 512
constexpr int   NROWS = B_ * T_ * U1_; // 133120
constexpr int   SLICE = 3264;          // floats per time slice (padded)
constexpr int   LPY_OFF = 520;         // B_*U1_
constexpr int   LD_OFF  = 1032;        // 520 + B_*U_
constexpr float SIGMA_  = 0.05f;

// Issue 16 async b32 copies: 32 lanes x 4B each = 128B/op, 2048B total
// (elements 0..511 of one logit row).  ASYNCcnt += 16.
__device__ __forceinline__ void async_row16(const float* gsrc, uint32_t lds0) {
#pragma unroll
  for (int k = 0; k < 16; ++k) {
    asm volatile("global_load_async_to_lds_b32 %0, %1, off"
                 :: "v"(lds0 + (uint32_t)(k * 128)), "v"(gsrc + k * 32)
                 : "memory");
  }
}

__global__ __launch_bounds__(256) void tdt_softmax_kernel(
    const float* __restrict__ label, const float* __restrict__ dur,
    const int* __restrict__ targets, float* __restrict__ comb) {
  __shared__ float buf[8][2][512];           // per-wave double buffer (32 KB)
  const int wid    = threadIdx.x >> 5;
  const int lane   = threadIdx.x & 31;
  const int gw     = blockIdx.x * 8 + wid;
  const int stride = gridDim.x * 8;

  const int row0 = gw;
  if (row0 < NROWS)
    async_row16(label + (size_t)row0 * VV + lane,
                (uint32_t)(uintptr_t)&buf[wid][0][lane]);

  int cur = 0;
  for (int row = row0; row < NROWS; row += stride) {
    const int nxt = row + stride;
    if (nxt < NROWS) {      // prefetch next row, then wait for current (<=16)
      async_row16(label + (size_t)nxt * VV + lane,
                  (uint32_t)(uintptr_t)&buf[wid][cur ^ 1][lane]);
      asm volatile("s_wait_asynccnt 16" ::: "memory");
    } else {
      asm volatile("s_wait_asynccnt 0" ::: "memory");
    }

    const float* base = label + (size_t)row * VV;
    const float  xb   = base[512];           // BLANK logit (uniform load)
    float vals[16];
    float m = xb;
#pragma unroll
    for (int k = 0; k < 16; ++k) {
      vals[k] = buf[wid][cur][k * 32 + lane];
      m = fmaxf(m, vals[k]);
    }
#pragma unroll
    for (int off = 16; off >= 1; off >>= 1)
      m = fmaxf(m, __shfl_xor(m, off, 32));
    float s = 0.f;
#pragma unroll
    for (int k = 0; k < 16; ++k) s += __expf(vals[k] - m);
#pragma unroll
    for (int off = 16; off >= 1; off >>= 1)
      s += __shfl_xor(s, off, 32);
    s += __expf(xb - m);
    const float lse = m + __logf(s);

    if (lane == 0) {
      const int b   = row / (T_ * U1_);
      const int rem = row - b * (T_ * U1_);
      const int t   = rem / U1_;
      const int u   = rem - t * U1_;
      const size_t sb = (size_t)t * SLICE;
      comb[sb + b * U1_ + u] = xb - lse - SIGMA_;          // lpb
      if (u < U_) {
        const int tgt = targets[b * U_ + u];               // tgt in [0,512)
        comb[sb + LPY_OFF + b * U_ + u] =
            buf[wid][cur][tgt] - lse - SIGMA_;             // lpy
      }
      const float4 dv = *(const float4*)(dur + (size_t)row * 4);
      const float md = fmaxf(fmaxf(dv.x, dv.y), fmaxf(dv.z, dv.w));
      const float sd = __expf(dv.x - md) + __expf(dv.y - md) +
                       __expf(dv.z - md) + __expf(dv.w - md);
      const float l2 = md + __logf(sd);
      float4 o;
      o.x = dv.x - l2; o.y = dv.y - l2; o.z = dv.z - l2; o.w = dv.w - l2;
      *(float4*)(comb + sb + LD_OFF + (size_t)(b * U1_ + u) * 4) = o;  // ld
    }
    cur ^= 1;
  }
}

__device__ __forceinline__ float lse4f(float a, float b, float c, float d) {
  const float m = fmaxf(fmaxf(a, b), fmaxf(c, d));
  return m + __logf(__expf(a - m) + __expf(b - m) +
                    __expf(c - m) + __expf(d - m));
}
__device__ __forceinline__ float laddf(float a, float b) {
  const float m = fmaxf(a, b), n = fminf(a, b);
  return m + __logf(1.f + __expf(n - m));
}

__global__ __launch_bounds__(96) void tdt_dp_kernel(
    const float* __restrict__ comb, float* __restrict__ alphas,
    const int* __restrict__ in_len, const int* __restrict__ tg_len,
    float* __restrict__ llbuf) {
  __shared__ float hist[4][U1_];        // alpha ring: slot (t&3) = alpha_t
  const int  b    = blockIdx.x;
  const int  u    = threadIdx.x;
  const bool act  = (u < U1_);
  const int  cidx = b * U1_ + u;

  // register pipelines: slot j holds time (t-1-j) during step t
  float p0=NEGF,p1=NEGF,p2=NEGF,p3=NEGF;   // lpb[col u]
  float q0=NEGF,q1=NEGF,q2=NEGF,q3=NEGF;   // lpy[col u-1]
  float4 nv = make_float4(NEGF, NEGF, NEGF, NEGF);
  float4 A0=nv,A1=nv,A2=nv,A3=nv;          // ld[col u]
  float4 Bm0=nv,Bm1=nv,Bm2=nv,Bm3=nv;      // ld[col u-1]
  float inp = NEGF, inq = NEGF;
  float4 inA = nv, inB = nv;

  if (act) {
    const float a0 = (u == 0) ? 0.f : NEGF;
    hist[0][u] = a0; hist[1][u] = NEGF; hist[2][u] = NEGF; hist[3][u] = NEGF;
    alphas[cidx] = a0;
    inp = comb[b * U1_ + u];                                   // time 0
    inA = *(const float4*)(comb + LD_OFF + (size_t)cidx * 4);
    if (u >= 1) {
      inq = comb[LPY_OFF + b * U_ + (u - 1)];
      inB = *(const float4*)(comb + LD_OFF + (size_t)(cidx - 1) * 4);
    }
  }
  __syncthreads();

  for (int t = 1; t < T_; ++t) {
    float alpha = NEGF;
    if (act) {
      // shift pipelines: slot0 becomes time t-1
      p3=p2; p2=p1; p1=p0; p0=inp;
      q3=q2; q2=q1; q1=q0; q0=inq;
      A3=A2; A2=A1; A1=A0; A0=inA;
      Bm3=Bm2; Bm2=Bm1; Bm1=Bm0; Bm0=inB;
      // prefetch time t (consumed next iteration -> latency hidden)
      const size_t sb = (size_t)t * SLICE;
      inp = comb[sb + b * U1_ + u];
      inA = *(const float4*)(comb + sb + LD_OFF + (size_t)cidx * 4);
      if (u >= 1) {
        inq = comb[sb + LPY_OFF + b * U_ + (u - 1)];
        inB = *(const float4*)(comb + sb + LD_OFF + (size_t)(cidx - 1) * 4);
      }
      const float h1 = hist[(t-1)&3][u], h2 = hist[(t-2)&3][u],
                  h3 = hist[(t-3)&3][u], h4 = hist[(t-4)&3][u];
      // duration component = d-1 for the time-(t-d) slot (slot d-1)
      const float blank_in =
          lse4f(h1 + p0 + A0.x, h2 + p1 + A1.y, h3 + p2 + A2.z, h4 + p3 + A3.w);
      if (u == 0) {
        alpha = blank_in;
      } else {
        const float g1 = hist[(t-1)&3][u-1], g2 = hist[(t-2)&3][u-1],
                    g3 = hist[(t-3)&3][u-1], g4 = hist[(t-4)&3][u-1];
        const float lab = lse4f(g1 + q0 + Bm0.x, g2 + q1 + Bm1.y,
                                g3 + q2 + Bm2.z, g4 + q3 + Bm3.w);
        alpha = laddf(blank_in, lab);
      }
    }
    __syncthreads();                       // all reads of slot t&3 done
    if (act) {
      hist[t & 3][u] = alpha;
      alphas[(size_t)t * (B_ * U1_) + cidx] = alpha;
    }
    __syncthreads();                       // writes visible for next step
  }

  __threadfence_block();
  __syncthreads();
  if (u == 0) {
    int Tl = in_len[b]; if (Tl > T_) Tl = T_;
    int Ul = tg_len[b]; if (Ul < 0) Ul = 0; if (Ul > U_) Ul = U_;
    float terms[4];
#pragma unroll
    for (int i = 0; i < 4; ++i) {
      const int ti = Tl - (i + 1);
      const int ts = ti < 0 ? 0 : (ti > T_ - 1 ? T_ - 1 : ti);
      const float a  = alphas[(size_t)ts * (B_ * U1_) + b * U1_ + Ul];
      const float lp = comb[(size_t)ts * SLICE + b * U1_ + Ul];
      const float ld = comb[(size_t)ts * SLICE + LD_OFF +
                            (size_t)(b * U1_ + Ul) * 4 + i];
      terms[i] = (ti >= 0) ? (a + lp + ld) : NEGF;
    }
    llbuf[b] = -lse4f(terms[0], terms[1], terms[2], terms[3]);
  }
}

__global__ void tdt_mean_kernel(const float* __restrict__ llbuf,
                                float* __restrict__ out) {
  if (threadIdx.x == 0 && blockIdx.x == 0) {
    float s = 0.f;
#pragma unroll
    for (int b = 0; b < B_; ++b) s += llbuf[b];
    out[0] = s / (float)B_;
  }
}

extern "C" void kernel_launch(void* const* d_in, const int* in_sizes, int n_in,
                              void* d_out, int out_size, void* d_ws,
                              size_t ws_size, hipStream_t stream) {
  (void)in_sizes; (void)n_in; (void)out_size; (void)ws_size;
  const float* label   = (const float*)d_in[0];  // (B,T,U+1,V+1) f32
  const float* dur     = (const float*)d_in[1];  // (B,T,U+1,D)   f32
  const int*   targets = (const int*)d_in[2];    // (B,U)         i32
  const int*   in_len  = (const int*)d_in[3];    // (B,)          i32
  const int*   tg_len  = (const int*)d_in[4];    // (B,)          i32

  float* ws     = (float*)d_ws;
  float* comb   = ws;                       // 256*3264 floats  (~3.3 MB)
  float* alphas = comb + (size_t)T_ * SLICE;       // 256*520 floats
  float* llbuf  = alphas + (size_t)T_ * B_ * U1_;  // 8 floats

  // Phase 1: 2080 blocks x 8 waves = 16640 waves, 8 rows/wave.
  tdt_softmax_kernel<<<2080, 256, 0, stream>>>(label, dur, targets, comb);
  // Phase 2: one block per batch element.
  tdt_dp_kernel<<<B_, 96, 0, stream>>>(comb, alphas, in_len, tg_len, llbuf);
  // Phase 3: scalar mean.
  tdt_mean_kernel<<<1, 32, 0, stream>>>(llbuf, (float*)d_out);
}